// GINEConv_69028714381386
// MI455X (gfx1250) — compile-verified
//
#include <hip/hip_runtime.h>

typedef __attribute__((ext_vector_type(2))) float v2f;
typedef __attribute__((ext_vector_type(8))) float v8f;

#define N_NODES 50000
#define N_EDGES 800000
#define NIN 96
#define NTILE_WAVES 6          // 6 waves x 16 cols = 96 output columns
#define BLOCK_THREADS (NTILE_WAVES * 32)
#define KSLICES (NIN / 4)      // 24 WMMA k-steps of 4 for f32 WMMA
#define EDGE_TILES (N_EDGES / 16)
#define NODE_TILES (N_NODES / 16)
#define LDS_STRIDE 100         // floats; 400B rows -> conflict-free & 8B aligned

// ---------------- zero workspace ----------------
__global__ void gine_zero_kernel(float4* __restrict__ p, int n4) {
    int i = blockIdx.x * blockDim.x + threadIdx.x;
    if (i < n4) p[i] = make_float4(0.f, 0.f, 0.f, 0.f);
}

// ---------------- edge kernel: ea = edge_attr @ We^T + be; msg = relu(x[src]+ea); atomic add to aggr[dst]
__global__ __launch_bounds__(BLOCK_THREADS) void gine_edge_kernel(
    const float* __restrict__ x,
    const unsigned* __restrict__ ei32,     // int64 edge_index viewed as u32 pairs (low word valid)
    const float* __restrict__ edge_attr,
    const float* __restrict__ We,
    const float* __restrict__ be,
    float* __restrict__ aggr)
{
    const int lane = threadIdx.x & 31;
    const int w    = threadIdx.x >> 5;
    const int hh   = lane >> 4;            // lane half
    const int nl   = lane & 15;
    const int n    = w * 16 + nl;          // output column owned by this lane

    // Preload B fragments of W_edge^T for this wave's 16 columns (L2-hot, regular hints).
    // B vreg j (j=0,1), half h: K = 4*ks + j + 2*h ; B[k][n] = We[n][k]
    v2f B[KSLICES];
#pragma unroll
    for (int ks = 0; ks < KSLICES; ++ks)
        B[ks] = *(const v2f*)(We + (size_t)n * NIN + ks * 4 + hh * 2);

    const float bias = be[n];

    for (int t = blockIdx.x; t < EDGE_TILES; t += gridDim.x) {
        const int e0 = t * 16;
        v8f c = {};
        const float* arow = edge_attr + (size_t)(e0 + nl) * NIN;
#pragma unroll
        for (int ks = 0; ks < KSLICES; ++ks) {
            // edge_attr streams through exactly once (307 MB): non-temporal so it
            // does not evict the L2-resident x/aggr node tables.
            v2f a = __builtin_nontemporal_load((const v2f*)(arow + ks * 4 + hh * 2));
            c = __builtin_amdgcn_wmma_f32_16x16x4_f32(
                    false, a, false, B[ks], (short)0, c, false, false);
        }
        // lanes 0-15 hold src[e0+nl], lanes 16-31 hold dst[e0+nl] (low 32 bits of int64)
        const unsigned e_l = (unsigned)(e0 + nl);
        int idx = (lane < 16)
                    ? (int)__builtin_nontemporal_load(ei32 + 2 * (size_t)e_l)
                    : (int)__builtin_nontemporal_load(ei32 + 2 * ((size_t)N_EDGES + e_l));
#pragma unroll
        for (int j = 0; j < 8; ++j) {
            const int m  = j + 8 * hh;                         // C/D row for vreg j
            const int sm = __builtin_amdgcn_ds_bpermute(4 * m,        idx);
            const int dm = __builtin_amdgcn_ds_bpermute(4 * (16 + m), idx);
            float v = c[j] + bias + x[(size_t)sm * NIN + n];   // x: hot, keep RT
            v = v > 0.f ? v : 0.f;
            __hip_atomic_fetch_add(&aggr[(size_t)dm * NIN + n], v,
                                   __ATOMIC_RELAXED, __HIP_MEMORY_SCOPE_AGENT);
        }
    }
}

// ---------------- node kernel: h=(1+eps)*x+aggr; h1=relu(h@W1^T+b1); out=h1@W2^T+b2
__global__ __launch_bounds__(BLOCK_THREADS) void gine_node_kernel(
    const float* __restrict__ x,
    const float* __restrict__ aggr,
    const float* __restrict__ eps_p,
    const float* __restrict__ W1, const float* __restrict__ b1,
    const float* __restrict__ W2, const float* __restrict__ b2,
    float* __restrict__ out)
{
    __shared__ __align__(16) float h1[16 * LDS_STRIDE];

    const int lane = threadIdx.x & 31;
    const int w    = threadIdx.x >> 5;
    const int hh   = lane >> 4;
    const int nl   = lane & 15;
    const int n    = w * 16 + nl;

    v2f B1[KSLICES], B2[KSLICES];
#pragma unroll
    for (int ks = 0; ks < KSLICES; ++ks) {
        B1[ks] = *(const v2f*)(W1 + (size_t)n * NIN + ks * 4 + hh * 2);
        B2[ks] = *(const v2f*)(W2 + (size_t)n * NIN + ks * 4 + hh * 2);
    }
    const float bias1 = b1[n];
    const float bias2 = b2[n];
    const float eps1  = 1.0f + eps_p[0];

    for (int t = blockIdx.x; t < NODE_TILES; t += gridDim.x) {
        const int r = t * 16 + nl;                 // A row (node) for this lane
        const float* xr = x    + (size_t)r * NIN;
        const float* ar = aggr + (size_t)r * NIN;

        v8f c = {};
#pragma unroll
        for (int ks = 0; ks < KSLICES; ++ks) {
            const int kb = ks * 4 + hh * 2;
            v2f xa = *(const v2f*)(xr + kb);
            v2f aa = *(const v2f*)(ar + kb);
            v2f a  = eps1 * xa + aa;               // h = (1+eps)*x + aggr, built on the fly
            c = __builtin_amdgcn_wmma_f32_16x16x4_f32(
                    false, a, false, B1[ks], (short)0, c, false, false);
        }

        __syncthreads();                           // prior-iteration LDS reads done
#pragma unroll
        for (int j = 0; j < 8; ++j) {
            const int m = j + 8 * hh;
            float v = c[j] + bias1;
            v = v > 0.f ? v : 0.f;
            h1[m * LDS_STRIDE + n] = v;            // conflict-free (stride 100)
        }
        __syncthreads();                           // full h1 tile visible

        v8f c2 = {};
#pragma unroll
        for (int ks = 0; ks < KSLICES; ++ks) {
            const int kb = ks * 4 + hh * 2;
            v2f a = *(const v2f*)&h1[nl * LDS_STRIDE + kb];  // ds_load_b64, conflict-free
            c2 = __builtin_amdgcn_wmma_f32_16x16x4_f32(
                     false, a, false, B2[ks], (short)0, c2, false, false);
        }
#pragma unroll
        for (int j = 0; j < 8; ++j) {
            const int m = j + 8 * hh;
            // output is write-once, never re-read: non-temporal store
            __builtin_nontemporal_store(c2[j] + bias2,
                                        out + (size_t)(t * 16 + m) * NIN + n);
        }
    }
}

extern "C" void kernel_launch(void* const* d_in, const int* in_sizes, int n_in,
                              void* d_out, int out_size, void* d_ws, size_t ws_size,
                              hipStream_t stream) {
    const float*    x     = (const float*)d_in[0];
    const unsigned* ei32  = (const unsigned*)d_in[1];   // int64 viewed as u32 pairs
    const float*    eattr = (const float*)d_in[2];
    const float*    We    = (const float*)d_in[3];
    const float*    be    = (const float*)d_in[4];
    const float*    eps   = (const float*)d_in[5];
    const float*    W1    = (const float*)d_in[6];
    const float*    b1    = (const float*)d_in[7];
    const float*    W2    = (const float*)d_in[8];
    const float*    b2    = (const float*)d_in[9];
    float*          out   = (float*)d_out;
    float*          aggr  = (float*)d_ws;              // 50000*96*4 = 19.2 MB

    // zero the aggregation buffer
    const int n4 = (N_NODES * NIN) / 4;
    gine_zero_kernel<<<(n4 + 255) / 256, 256, 0, stream>>>((float4*)aggr, n4);

    // edge pipeline: WMMA edge-GEMM + gather/ReLU + atomic scatter-add
    gine_edge_kernel<<<2500, BLOCK_THREADS, 0, stream>>>(x, ei32, eattr, We, be, aggr);

    // node MLP: two chained WMMA GEMMs through LDS
    gine_node_kernel<<<625, BLOCK_THREADS, 0, stream>>>(x, aggr, eps, W1, b1, W2, b2, out);
}